// GHMCLoss_11123965296940
// MI455X (gfx1250) — compile-verified
//
#include <hip/hip_runtime.h>
#include <hip/hip_bf16.h>
#include <stdint.h>

#define BINS 30
#define TPB  256

// Low 32 bits of a generic pointer into LDS are the LDS byte offset
// (aperture lives in addr[63:32] per CDNA5 aperture rules).
__device__ __forceinline__ unsigned lds_off(const void* p) {
    return (unsigned)(uintptr_t)p;
}

// ---------------------------------------------------------------------------
// Kernel 1: single fused pass. Async-stage float4 tiles of preds/targets into
// double-buffered LDS, compute g/bin/bce per element, accumulate per-bin
// count + bce-sum with ds_add_f32 atomics, write per-block partials to ws.
// ---------------------------------------------------------------------------
__global__ __launch_bounds__(TPB) void ghmc_hist_kernel(
        const float* __restrict__ preds,
        const float* __restrict__ targets,
        float* __restrict__ ws,
        int numTiles, long long N)
{
    __shared__ __align__(16) float buf[2][2][TPB * 4];  // [dbl-buf][preds/targets][1024 f32] = 16 KB
    __shared__ float shc[32];   // per-bin counts (exact in f32)
    __shared__ float shs[32];   // per-bin bce sums

    const int tid = threadIdx.x;
    if (tid < 32) { shc[tid] = 0.0f; shs[tid] = 0.0f; }
    __syncthreads();

    const unsigned ldsP0 = lds_off(&buf[0][0][tid * 4]);
    const unsigned ldsT0 = lds_off(&buf[0][1][tid * 4]);
    const unsigned BUFSTRIDE = 2u * TPB * 4u * (unsigned)sizeof(float);  // 8192 B

    auto elem = [&](float x, float t) {
        float ax  = fabsf(x);
        float e   = __expf(-ax);                      // v_exp_f32 path
        float r   = __builtin_amdgcn_rcpf(1.0f + e);  // v_rcp_f32
        float sig = (x >= 0.0f) ? r : e * r;          // sigmoid(x), one exp total
        float g   = fabsf(sig - t);
        int bi = (int)(g * (float)BINS);
        bi = bi > (BINS - 1) ? (BINS - 1) : bi;
        float bce = fmaxf(x, 0.0f) - x * t + __logf(1.0f + e);  // stable BCE
        atomicAdd(&shc[bi], 1.0f);   // ds_add_f32 (no-return)
        atomicAdd(&shs[bi], bce);    // ds_add_f32 (no-return)
    };

    auto issue = [&](int tile, int b) {
        unsigned goff = ((unsigned)tile * TPB + (unsigned)tid) * 16u;  // byte offset, < 2^31
        unsigned lp = ldsP0 + (unsigned)b * BUFSTRIDE;
        unsigned lt = ldsT0 + (unsigned)b * BUFSTRIDE;
        asm volatile("global_load_async_to_lds_b128 %0, %1, %2 offset:0"
                     :: "v"(lp), "v"(goff), "s"(preds) : "memory");
        asm volatile("global_load_async_to_lds_b128 %0, %1, %2 offset:0"
                     :: "v"(lt), "v"(goff), "s"(targets) : "memory");
    };

    auto process = [&](int b) {
        const float4 p4 = *(const float4*)&buf[b][0][tid * 4];
        const float4 t4 = *(const float4*)&buf[b][1][tid * 4];
        elem(p4.x, t4.x); elem(p4.y, t4.y);
        elem(p4.z, t4.z); elem(p4.w, t4.w);
    };

    // Double-buffered grid-stride loop over 4 KB-per-array tiles.
    int tile = blockIdx.x;
    int bsel = 0;
    if (tile < numTiles) issue(tile, bsel);
    for (; tile < numTiles; tile += gridDim.x) {
        int next = tile + (int)gridDim.x;
        if (next < numTiles) {                 // uniform branch per block
            issue(next, bsel ^ 1);
            // 2 newer async ops outstanding; in-order completion =>
            // current tile's 2 loads have landed in LDS.
            asm volatile("s_wait_asynccnt 2" ::: "memory");
        } else {
            asm volatile("s_wait_asynccnt 0" ::: "memory");
        }
        process(bsel);
        bsel ^= 1;
    }

    // Generic tails (empty for N = 16,384,000 but kept for safety).
    long long N4 = N >> 2;
    long long stride = (long long)gridDim.x * TPB;
    for (long long i = (long long)numTiles * TPB + (long long)blockIdx.x * TPB + tid;
         i < N4; i += stride) {
        const float4 p4 = ((const float4*)preds)[i];
        const float4 t4 = ((const float4*)targets)[i];
        elem(p4.x, t4.x); elem(p4.y, t4.y); elem(p4.z, t4.z); elem(p4.w, t4.w);
    }
    for (long long i = (N4 << 2) + (long long)blockIdx.x * TPB + tid; i < N; i += stride) {
        elem(preds[i], targets[i]);
    }

    __syncthreads();
    if (tid < 32) {   // deterministic per-block partials: [counts(32) | bce_sums(32)]
        ws[(long long)blockIdx.x * 64 + tid]      = shc[tid];
        ws[(long long)blockIdx.x * 64 + 32 + tid] = shs[tid];
    }
}

// ---------------------------------------------------------------------------
// Kernel 2: reduce per-block partials, momentum update, weight, emit scalar.
// ---------------------------------------------------------------------------
__global__ __launch_bounds__(256) void ghmc_final_kernel(
        const float* __restrict__ ws,
        const float* __restrict__ acc_sum,
        float* __restrict__ out,
        int nblocks, float tot)
{
    const int tid  = threadIdx.x;
    const int lane = tid & 31;
    const int grp  = tid >> 5;          // 8 groups of one wave32 each
    float c = 0.0f, s = 0.0f;
    if (lane < BINS) {
        for (int b = grp; b < nblocks; b += 8) {
            c += ws[(long long)b * 64 + lane];
            s += ws[(long long)b * 64 + 32 + lane];
        }
    }
    __shared__ float shc[8][32];
    __shared__ float shs[8][32];
    shc[grp][lane] = c;
    shs[grp][lane] = s;
    __syncthreads();

    if (tid < 32) {                     // one wave: lane b owns bin b
        float C = 0.0f, S = 0.0f;
        #pragma unroll
        for (int g = 0; g < 8; ++g) { C += shc[g][tid]; S += shs[g][tid]; }
        bool valid = tid < BINS;
        bool nz    = valid && (C > 0.0f);
        float acc  = valid ? acc_sum[tid] : 0.0f;
        float nacc = nz ? (0.75f * acc + 0.25f * C) : acc;   // MOMENTUM = 0.75
        float contrib = nz ? (tot / nacc) * S : 0.0f;        // per_bin_w * bce_sum
        float nf      = nz ? 1.0f : 0.0f;
        #pragma unroll
        for (int m = 16; m >= 1; m >>= 1) {                  // wave32 tree reduce
            contrib += __shfl_xor(contrib, m, 32);
            nf      += __shfl_xor(nf, m, 32);
        }
        if (tid == 0) {
            float n = fmaxf(nf, 1.0f);
            out[0] = (contrib / (n * tot)) * 1.0f;           // LOSS_WEIGHT = 1.0
        }
    }
}

extern "C" void kernel_launch(void* const* d_in, const int* in_sizes, int n_in,
                              void* d_out, int out_size, void* d_ws, size_t ws_size,
                              hipStream_t stream) {
    const float* preds   = (const float*)d_in[0];
    const float* targets = (const float*)d_in[1];
    const float* acc_sum = (const float*)d_in[2];
    float* out = (float*)d_out;
    float* ws  = (float*)d_ws;

    long long N  = (long long)in_sizes[0];   // 16,384,000
    long long N4 = N >> 2;
    int numTiles = (int)(N4 / TPB);          // 16000 full tiles

    int nblocks = 2048;                      // 524k threads; ~8 tiles/block
    size_t need = (size_t)nblocks * 64 * sizeof(float);
    if (need > ws_size) {
        nblocks = (int)(ws_size / (64 * sizeof(float)));
        if (nblocks < 1) nblocks = 1;
    }

    ghmc_hist_kernel<<<nblocks, TPB, 0, stream>>>(preds, targets, ws, numTiles, N);
    ghmc_final_kernel<<<1, 256, 0, stream>>>(ws, acc_sum, out, nblocks, (float)N);
}